// BertNormOutput_54726473285725
// MI455X (gfx1250) — compile-verified
//
#include <hip/hip_runtime.h>
#include <hip/hip_bf16.h>

// Problem constants (from the reference)
#define BB 2
#define HH 12
#define SS 512
#define DD 64
#define AA 768   // H*D

typedef __attribute__((ext_vector_type(2))) float v2f;
typedef __attribute__((ext_vector_type(8))) float v8f;

// ---------------------------------------------------------------------------
// Kernel 1: transformed[b,h,s,e] = sum_d V[b,h,s,d] * W[e, h*64+d]
// Per (b,h): C[512x768] = A[512x64] * Bm[64x768] with
//   A[m,k]  = V[((b*H+h)*S + m)*64 + k]          (row-major, stride 64)
//   Bm[k,n] = W[n*768 + h*64 + k]                (column-major slice of W)
// One wave computes one 16x16 C tile via 16x v_wmma_f32_16x16x4_f32.
// Fragment layouts per CDNA5 ISA 7.12.2:
//   A 16x4 f32 : lanes 0-15 -> M=lane, VGPR0=K0,VGPR1=K1 ; lanes16-31 -> K2,K3
//   B 4x16 f32 : lanes 0-15 -> N=lane, VGPR0=K0,VGPR1=K1 ; lanes16-31 -> K2,K3
//   C/D 16x16  : VGPR v     -> lanes 0-15 row v, lanes 16-31 row v+8
// ---------------------------------------------------------------------------
__global__ __launch_bounds__(256) void k1_wmma_gemm(
    const float* __restrict__ V,   // [B,H,S,D]
    const float* __restrict__ W,   // [ALL,ALL]
    float* __restrict__ T)         // [B,H,S,ALL]
{
    const int TN = AA / 16;            // 48 tiles along e
    const int TM = SS / 16;            // 32 tiles along s
    const int wave = threadIdx.x >> 5; // 8 waves per block
    const int lane = threadIdx.x & 31;

    long tile = (long)blockIdx.x * 8 + wave;   // total tiles = 24*32*48, /8 blocks
    int n_t = (int)(tile % TN);
    int m_t = (int)((tile / TN) % TM);
    int bh  = (int)(tile / ((long)TN * TM));   // b*H + h
    int h   = bh % HH;

    const float* Vbh = V + (long)bh * SS * DD;
    const float* Wh  = W + h * DD;             // Wh[k,n] = Wh[n*AA + k]
    float*       Tbh = T + (long)bh * SS * AA;

    const int m0 = m_t * 16;
    const int n0 = n_t * 16;
    const int half = lane >> 4;                // 0: K pair {0,1}; 1: {2,3}
    const int l    = lane & 15;

    v8f acc = {};
    #pragma unroll
    for (int k0 = 0; k0 < DD; k0 += 4) {
        const int ka = k0 + 2 * half;
        v2f a, b;
        const float* ap = Vbh + (long)(m0 + l) * DD + ka;
        a.x = ap[0];
        a.y = ap[1];
        const float* bp = Wh + (long)(n0 + l) * AA + ka;
        b.x = bp[0];
        b.y = bp[1];
        acc = __builtin_amdgcn_wmma_f32_16x16x4_f32(
            /*neg_a=*/false, a, /*neg_b=*/false, b,
            /*c_mod=*/(short)0, acc, /*reuse_a=*/false, /*reuse_b=*/false);
    }

    #pragma unroll
    for (int v = 0; v < 8; ++v) {
        const int row = m0 + v + 8 * half;
        Tbh[(long)row * AA + (n0 + l)] = acc[v];
    }
}

// ---------------------------------------------------------------------------
// Kernel 2: per (b,s) compute 12x12 head Gram  G[h,h'] = sum_e T[b,h,s,e]T[b,h',s,e]
// and transformed_norm[b,h,s] = sqrt(G[h,h]).  T rows staged in LDS (36 KB).
// ---------------------------------------------------------------------------
__global__ __launch_bounds__(256) void k2_gram_norm(
    const float* __restrict__ T,   // [B,H,S,ALL]
    float* __restrict__ tn,        // [B,H,S]
    float* __restrict__ G)         // [B,S,144]
{
    const int bs = blockIdx.x;         // 0..B*S-1
    const int b  = bs / SS;
    const int s  = bs % SS;

    __shared__ float sh[HH * AA];      // 12*768*4 = 36864 B
    for (int i = threadIdx.x; i < HH * AA; i += 256) {
        const int h = i / AA, e = i % AA;
        sh[i] = T[(((long)b * HH + h) * SS + s) * AA + e];
    }
    __syncthreads();

    const int t = threadIdx.x;
    if (t < 144) {
        const int h  = t / HH;
        const int hp = t % HH;
        const float* x = sh + h * AA;
        const float* y = sh + hp * AA;
        float acc = 0.f;
        #pragma unroll 8
        for (int e = 0; e < AA; ++e) acc = fmaf(x[e], y[e], acc);
        G[(long)bs * 144 + t] = acc;
        if (h == hp) tn[((long)b * HH + h) * SS + s] = sqrtf(acc);
    }
}

// ---------------------------------------------------------------------------
// Kernel 3: weighted_norm[b,h,k,s] = |P[b,h,k,s]| * transformed_norm[b,h,s]
// ---------------------------------------------------------------------------
__global__ __launch_bounds__(256) void k3_weighted(
    const float* __restrict__ P,   // [B,H,S,S]
    const float* __restrict__ tn,  // [B,H,S]
    float* __restrict__ wn)        // [B,H,S,S]
{
    const long n = (long)BB * HH * SS * SS;
    long i = (long)blockIdx.x * 256 + threadIdx.x;
    if (i >= n) return;
    const int  s  = (int)(i % SS);
    const long bh = i / ((long)SS * SS);   // b*H + h
    wn[i] = fabsf(P[i]) * tn[bh * SS + s];
}

// ---------------------------------------------------------------------------
// Kernel 4: summed_weighted_norm[b,k,s] = sqrt( p^T G[b,s] p ),  p = P[b,:,k,s]
// Workgroup handles (b, 64-k chunk, 64-s tile); Gram rows staged in LDS.
// Consecutive threads -> consecutive s -> coalesced P reads.
// ---------------------------------------------------------------------------
__global__ __launch_bounds__(256) void k4_qform(
    const float* __restrict__ P,   // [B,H,S,S]
    const float* __restrict__ G,   // [B,S,144]
    float* __restrict__ out)       // [B,S,S]  (k major, s minor)
{
    const int S_TILE = 64, K_CHUNK = 64;
    const int nst = SS / S_TILE;       // 8
    const int nkc = SS / K_CHUNK;      // 8
    const int x  = blockIdx.x;         // B*8*8 = 128 blocks
    const int st = x % nst;
    const int kc = (x / nst) % nkc;
    const int b  = x / (nst * nkc);
    const int s0 = st * S_TILE;
    const int k0 = kc * K_CHUNK;

    __shared__ float gsh[S_TILE * 144];    // 36864 B
    for (int i = threadIdx.x; i < S_TILE * 144; i += 256)
        gsh[i] = G[((long)b * SS + s0 + i / 144) * 144 + (i % 144)];
    __syncthreads();

    for (int idx = threadIdx.x; idx < K_CHUNK * S_TILE; idx += 256) {
        const int si = idx % S_TILE;
        const int s  = s0 + si;
        const int k  = k0 + idx / S_TILE;

        float p[HH];
        #pragma unroll
        for (int h = 0; h < HH; ++h)
            p[h] = P[(((long)b * HH + h) * SS + k) * SS + s];

        const float* g = gsh + si * 144;
        float q = 0.f;
        #pragma unroll
        for (int h = 0; h < HH; ++h) {
            float m = 0.f;
            #pragma unroll
            for (int hp = 0; hp < HH; ++hp) m = fmaf(g[h * HH + hp], p[hp], m);
            q = fmaf(p[h], m, q);
        }
        out[((long)b * SS + k) * SS + s] = sqrtf(fmaxf(q, 0.f));
    }
}

// ---------------------------------------------------------------------------
// Launch: inputs = {hidden_states(unused), attention_probs, value_layer, dense_weight}
// outputs concatenated: transformed_norm | weighted_norm | summed_weighted_norm | transformed
// ---------------------------------------------------------------------------
extern "C" void kernel_launch(void* const* d_in, const int* in_sizes, int n_in,
                              void* d_out, int out_size, void* d_ws, size_t ws_size,
                              hipStream_t stream) {
    (void)in_sizes; (void)n_in; (void)out_size; (void)ws_size;

    const float* P = (const float*)d_in[1];   // [B,H,S,S]
    const float* V = (const float*)d_in[2];   // [B,H,S,D]
    const float* W = (const float*)d_in[3];   // [ALL,ALL]

    float* outf = (float*)d_out;
    float* tn   = outf;                                         // B*H*S      = 12288
    float* wn   = tn + (long)BB * HH * SS;                      // B*H*S*S    = 6291456
    float* swn  = wn + (long)BB * HH * SS * SS;                 // B*S*S      = 524288
    float* T    = swn + (long)BB * SS * SS;                     // B*H*S*ALL  = 9437184

    float* G = (float*)d_ws;                                    // B*S*144 floats = 576 KB

    // K1: 24 (b,h) GEMMs, 32x48 tiles each, 8 tiles (waves) per block
    {
        const long tiles = (long)BB * HH * (SS / 16) * (AA / 16);   // 36864
        k1_wmma_gemm<<<(int)(tiles / 8), 256, 0, stream>>>(V, W, T);
    }
    // K2: one block per (b,s)
    k2_gram_norm<<<BB * SS, 256, 0, stream>>>(T, tn, G);
    // K3: elementwise over B*H*S*S
    {
        const long n = (long)BB * HH * SS * SS;
        k3_weighted<<<(int)((n + 255) / 256), 256, 0, stream>>>(P, tn, wn);
    }
    // K4: 128 blocks
    k4_qform<<<BB * (SS / 64) * (SS / 64), 256, 0, stream>>>(P, G, swn);
}